// SpAttentionLayer_17171279249899
// MI455X (gfx1250) — compile-verified
//
#include <hip/hip_runtime.h>
#include <hip/hip_bf16.h>

typedef __attribute__((ext_vector_type(2))) float v2f;
typedef __attribute__((ext_vector_type(8))) float v8f;

#define IN_FEAT 128
#define OUT_FEAT 64
#define ALPHA 0.2f

// ---------------------------------------------------------------- zero accums
__global__ void gat_zero_kernel(float* __restrict__ out,
                                float* __restrict__ rowsum, int n_nodes) {
    int idx = blockIdx.x * blockDim.x + threadIdx.x;
    if (idx < n_nodes * OUT_FEAT) out[idx] = 0.0f;
    if (idx < n_nodes) rowsum[idx] = 0.0f;
}

// ------------------------------------------------- h = x @ W  (WMMA f32 16x16x4)
// One wave32 per 16-row stripe of h. A frag: lanes0-15 hold row M=lane, K=kk..kk+1
// (kk = 0 for lanes 0-15, 2 for lanes 16-31). B frag: lane&15 = column N, same K
// split. C/D: VGPR i = rows i (lanes 0-15) and i+8 (lanes 16-31), N = lane&15.
__global__ void gat_gemm_wmma_kernel(const float* __restrict__ x,
                                     const float* __restrict__ W,
                                     float* __restrict__ h, int n_nodes) {
    const int lane  = threadIdx.x & 31;
    const int wave  = (blockIdx.x * blockDim.x + threadIdx.x) >> 5;
    const int ntile = (n_nodes + 15) >> 4;
    if (wave >= ntile) return;
    const int row0 = wave << 4;

    const int m  = lane & 15;
    const int kk = (lane >> 4) << 1;              // 0 or 2
    int arow = row0 + m;
    if (arow >= n_nodes) arow = n_nodes - 1;      // clamp loads; stores predicated
    const float* __restrict__ xrow = x + (size_t)arow * IN_FEAT;

    v8f acc0 = {}, acc1 = {}, acc2 = {}, acc3 = {};

#pragma unroll
    for (int k0 = 0; k0 < IN_FEAT; k0 += 4) {
        v2f a;
        a.x = xrow[k0 + kk];
        a.y = xrow[k0 + kk + 1];
        const float* __restrict__ w0 = W + (size_t)(k0 + kk) * OUT_FEAT + m;
        const float* __restrict__ w1 = w0 + OUT_FEAT;
        v2f b;
        b.x = w0[0];   b.y = w1[0];
        acc0 = __builtin_amdgcn_wmma_f32_16x16x4_f32(false, a, false, b,
                                                     (short)0, acc0, false, false);
        b.x = w0[16];  b.y = w1[16];
        acc1 = __builtin_amdgcn_wmma_f32_16x16x4_f32(false, a, false, b,
                                                     (short)0, acc1, false, false);
        b.x = w0[32];  b.y = w1[32];
        acc2 = __builtin_amdgcn_wmma_f32_16x16x4_f32(false, a, false, b,
                                                     (short)0, acc2, false, false);
        b.x = w0[48];  b.y = w1[48];
        acc3 = __builtin_amdgcn_wmma_f32_16x16x4_f32(false, a, false, b,
                                                     (short)0, acc3, false, false);
    }

    const int mbase = (lane >> 4) << 3;           // 0 or 8
    const int n     = lane & 15;
#pragma unroll
    for (int i = 0; i < 8; ++i) {
        const int row = row0 + mbase + i;
        if (row < n_nodes) {
            float* __restrict__ hr = h + (size_t)row * OUT_FEAT + n;
            hr[0]  = acc0[i];
            hr[16] = acc1[i];
            hr[32] = acc2[i];
            hr[48] = acc3[i];
        }
    }
}

// ------------------------------------------- s1 = h@a1, s2 = h@a2 (wave per node)
__global__ void gat_scores_kernel(const float* __restrict__ h,
                                  const float* __restrict__ a,
                                  float* __restrict__ s1,
                                  float* __restrict__ s2, int n_nodes) {
    const int lane = threadIdx.x & 31;
    const int node = (blockIdx.x * blockDim.x + threadIdx.x) >> 5;
    if (node >= n_nodes) return;
    const float h0 = h[(size_t)node * OUT_FEAT + lane];
    const float h1 = h[(size_t)node * OUT_FEAT + lane + 32];
    float p1 = h0 * a[lane]      + h1 * a[lane + 32];        // a1 = a[0:64]
    float p2 = h0 * a[lane + 64] + h1 * a[lane + 96];        // a2 = a[64:128]
#pragma unroll
    for (int off = 16; off >= 1; off >>= 1) {
        p1 += __shfl_xor(p1, off, 32);
        p2 += __shfl_xor(p2, off, 32);
    }
    if (lane == 0) { s1[node] = p1; s2[node] = p2; }
}

// ------------------------- edge_e = exp(-leaky_relu(score)); rowsum[src] += edge_e
__global__ void gat_edge_kernel(const int* __restrict__ src,
                                const int* __restrict__ dst,
                                const float* __restrict__ s1,
                                const float* __restrict__ s2,
                                float* __restrict__ edge_e,
                                float* __restrict__ rowsum, int n_edges) {
    int e = blockIdx.x * blockDim.x + threadIdx.x;
    if (e >= n_edges) return;
    const int s = src[e];
    const int d = dst[e];
    const float sc = s1[s] + s2[d];
    const float lr = sc > 0.0f ? sc : ALPHA * sc;
    const float ee = expf(-lr);
    edge_e[e] = ee;
    atomicAdd(&rowsum[s], ee);
}

// ------------------------------ out[src] += edge_e * h[dst]   (wave32 per edge)
__global__ void gat_scatter_kernel(const int* __restrict__ src,
                                   const int* __restrict__ dst,
                                   const float* __restrict__ edge_e,
                                   const float* __restrict__ h,
                                   float* __restrict__ out, int n_edges) {
    const int lane = threadIdx.x & 31;
    const int e    = (blockIdx.x * blockDim.x + threadIdx.x) >> 5;
    if (e >= n_edges) return;
    const int s = src[e];
    const int d = dst[e];
    const float ee = edge_e[e];
    const float* __restrict__ hd = h + (size_t)d * OUT_FEAT;
    float* __restrict__ os = out + (size_t)s * OUT_FEAT;
    atomicAdd(&os[lane],      ee * hd[lane]);
    atomicAdd(&os[lane + 32], ee * hd[lane + 32]);
}

// ------------------------------------------- out = elu(out / rowsum), in place
__global__ void gat_finalize_kernel(float* __restrict__ out,
                                    const float* __restrict__ rowsum,
                                    int n_nodes) {
    int idx = blockIdx.x * blockDim.x + threadIdx.x;
    if (idx >= n_nodes * OUT_FEAT) return;
    const float v = out[idx] / rowsum[idx >> 6];
    out[idx] = v > 0.0f ? v : (expf(v) - 1.0f);
}

// -----------------------------------------------------------------------------
extern "C" void kernel_launch(void* const* d_in, const int* in_sizes, int n_in,
                              void* d_out, int out_size, void* d_ws, size_t ws_size,
                              hipStream_t stream) {
    const float* x    = (const float*)d_in[0];   // [N, 128]
    const float* W    = (const float*)d_in[1];   // [128, 64]
    const float* a    = (const float*)d_in[2];   // [1, 128]
    const int*   edge = (const int*)d_in[3];     // [2, E]

    const int n_nodes = in_sizes[0] / IN_FEAT;
    const int n_edges = in_sizes[3] / 2;
    const int* src = edge;
    const int* dst = edge + n_edges;
    float* out = (float*)d_out;

    char* ws = (char*)d_ws;
    float* h      = (float*)ws;  ws += (size_t)n_nodes * OUT_FEAT * sizeof(float);
    float* s1     = (float*)ws;  ws += (size_t)n_nodes * sizeof(float);
    float* s2     = (float*)ws;  ws += (size_t)n_nodes * sizeof(float);
    float* rowsum = (float*)ws;  ws += (size_t)n_nodes * sizeof(float);
    float* edge_e = (float*)ws;  // n_edges floats

    // 0) zero accumulators (required every call: graph replay reuses buffers)
    gat_zero_kernel<<<(n_nodes * OUT_FEAT + 255) / 256, 256, 0, stream>>>(
        out, rowsum, n_nodes);

    // 1) h = x @ W  (WMMA; one wave per 16 rows, 8 waves/block)
    const int ntile = (n_nodes + 15) / 16;
    gat_gemm_wmma_kernel<<<(ntile + 7) / 8, 256, 0, stream>>>(x, W, h, n_nodes);

    // 2) per-node attention scores (wave per node)
    gat_scores_kernel<<<(n_nodes + 7) / 8, 256, 0, stream>>>(h, a, s1, s2, n_nodes);

    // 3) per-edge coefficients + rowsum
    gat_edge_kernel<<<(n_edges + 255) / 256, 256, 0, stream>>>(
        src, dst, s1, s2, edge_e, rowsum, n_edges);

    // 4) weighted scatter (wave per edge, 2 features per lane)
    gat_scatter_kernel<<<(n_edges + 7) / 8, 256, 0, stream>>>(
        src, dst, edge_e, h, out, n_edges);

    // 5) normalize + ELU
    gat_finalize_kernel<<<(n_nodes * OUT_FEAT + 255) / 256, 256, 0, stream>>>(
        out, rowsum, n_nodes);
}